// GNN_69483980915421
// MI455X (gfx1250) — compile-verified
//
#include <hip/hip_runtime.h>
#include <hip/hip_bf16.h>

// ---------------------------------------------------------------------------
// GAT-like reference collapses algebraically:
//   softmax_j(s1_i + s2_j) == softmax_j(s2_j)   (exact; shift invariance)
// => attention output is rank-1 in i, and the whole op becomes:
//   u_h   = W_heads[h] @ Wa2[h]                 (896-vec, K0)
//   s2    = xr @ U                              (streams x, WMMA, K1)
//   w     = softmax_j(s2)                       (K2)
//   y     = w^T @ xr (per b, per head)          (streams x from L2, WMMA, K3)
//   v     = y @ W_heads + B_gnn                 (tiny, K4)
//   out[b,t,c,f] = v[b, t*128+f]  (broadcast over c)   (117MB store, K5)
// Bandwidth-bound: ~240MB HBM @ 23.3 TB/s ~= 10us. fp32 WMMA (16x16x4) keeps
// numerics exact; padded-N/M compute hides under the memory stream.
// ---------------------------------------------------------------------------

typedef __attribute__((ext_vector_type(2))) float v2f;
typedef __attribute__((ext_vector_type(8))) float v8f;

#define B_   64
#define T_   7
#define C_   512
#define F_   128
#define D_   896     // T*F
#define HD_  224
#define G_   896     // 4*HD

// workspace layout (float offsets)
#define WS_U   0          // [896][16]   U padded to 16 head-cols
#define WS_S   14336      // [64][4][512] raw scores s2
#define WS_W   145408     // [64][4][512] softmax weights
#define WS_Y   276480     // [64][4][896] y vectors
#define WS_VB  505856     // [64][896]    v + bias
// total = 563200 floats = 2,252,800 bytes

// ---------------- K0: U[d][n] = sum_e W_heads[n][d][e] * Wa[n][HD+e] --------
__global__ void __launch_bounds__(256)
k0_u(const float* __restrict__ Wh, const float* __restrict__ Wa,
     float* __restrict__ U)
{
    int idx = blockIdx.x * 256 + threadIdx.x;   // 0..14335
    int d = idx >> 4, n = idx & 15;
    float acc = 0.f;
    if (n < 4) {
        const float* w = Wh + ((size_t)n * D_ + d) * HD_;
        const float* a = Wa + n * (2 * HD_) + HD_;
        #pragma unroll 4
        for (int e = 0; e < HD_; ++e) acc += w[e] * a[e];
    }
    U[idx] = acc;
}

// ---------------- K1: S[b][h][j] = xr[b,j,:] . u_h   (WMMA, streams x) ------
// grid: 512 = 64 b * 8 jblocks(64 j); block: 128 (4 waves, 16 j each)
__global__ void __launch_bounds__(128)
k1_scores(const float* __restrict__ x, const float* __restrict__ U,
          float* __restrict__ S)
{
    __shared__ float xs[64 * 132];   // 64 j rows, 128 d cols, pad->132 (16B-aligned rows)
    __shared__ float ul[128 * 16];   // U chunk: 128 d rows x 16 head cols

    const int b    = blockIdx.x >> 3;
    const int jblk = blockIdx.x & 7;
    const int tid  = threadIdx.x;
    const int wv   = tid >> 5;
    const int lane = tid & 31;
    const int half = lane >> 4;
    const int m    = lane & 15;

    v8f acc = {};
    const size_t xbase = (size_t)b * T_ * C_ * F_ + (size_t)(jblk * 64) * F_;

    for (int kc = 0; kc < 7; ++kc) {            // K chunks of 128 == one t slab
        const float4* src = (const float4*)(x + xbase + (size_t)kc * C_ * F_);
        #pragma unroll
        for (int i = 0; i < 16; ++i) {          // 2048 float4
            int idx = tid + i * 128;
            int jj = idx >> 5, q = idx & 31;
            *(float4*)&xs[jj * 132 + q * 4] = src[(size_t)jj * 32 + q];
        }
        #pragma unroll
        for (int i = 0; i < 4; ++i) {           // 512 float4 of U chunk
            int idx = tid + i * 128;
            *(float4*)&ul[idx * 4] = *(const float4*)&U[kc * 128 * 16 + idx * 4];
        }
        __syncthreads();

        const int jrow = (wv * 16 + m) * 132;
        #pragma unroll 8
        for (int k4 = 0; k4 < 32; ++k4) {
            int k0 = k4 * 4 + 2 * half;
            v2f a, bf;
            a.x  = xs[jrow + k0];               // A[m][k0],   A 16x4 layout
            a.y  = xs[jrow + k0 + 1];           // A[m][k0+1]
            bf.x = ul[k0 * 16 + m];             // B[k0][m],   B 4x16 layout
            bf.y = ul[(k0 + 1) * 16 + m];       // B[k0+1][m]
            acc = __builtin_amdgcn_wmma_f32_16x16x4_f32(
                      false, a, false, bf, (short)0, acc, false, false);
        }
        __syncthreads();
    }

    // C layout: VGPR r, lane -> (M = r + 8*half, N = m); keep head cols N<4
    if (m < 4) {
        int j0 = jblk * 64 + wv * 16 + 8 * half;
        float* Sp = S + ((size_t)b * 4 + m) * C_ + j0;
        #pragma unroll
        for (int r = 0; r < 8; ++r) Sp[r] = acc[r];
    }
}

// ---------------- K2: softmax over j for each (b,h) -------------------------
// grid: 32 blocks * 8 waves = 256 (b,h) pairs
__global__ void __launch_bounds__(256)
k2_softmax(const float* __restrict__ S, float* __restrict__ W)
{
    const int wv = threadIdx.x >> 5, lane = threadIdx.x & 31;
    const int p  = blockIdx.x * 8 + wv;         // b*4 + h
    const float* Sp = S + (size_t)p * C_;
    float*       Wp = W + (size_t)p * C_;

    float v[16];
    float mx = -3.4e38f;
    #pragma unroll
    for (int i = 0; i < 16; ++i) { v[i] = Sp[lane + i * 32]; mx = fmaxf(mx, v[i]); }
    #pragma unroll
    for (int o = 16; o > 0; o >>= 1) mx = fmaxf(mx, __shfl_xor(mx, o));
    float sum = 0.f;
    #pragma unroll
    for (int i = 0; i < 16; ++i) { v[i] = __expf(v[i] - mx); sum += v[i]; }
    #pragma unroll
    for (int o = 16; o > 0; o >>= 1) sum += __shfl_xor(sum, o);
    const float inv = 1.f / sum;
    #pragma unroll
    for (int i = 0; i < 16; ++i) Wp[lane + i * 32] = v[i] * inv;
}

// ---------------- K3: Y[b][h][d] = sum_j w[b,h,j] * xr[b,j,d]  (WMMA) -------
// grid: 448 = 64 b * 7 d-chunks(128); block 256 (8 waves, 16 d-cols each)
__global__ void __launch_bounds__(256)
k3_y(const float* __restrict__ x, const float* __restrict__ W,
     float* __restrict__ Y)
{
    __shared__ float wt[512 * 16];   // wT[j][16]: cols 0..3 = heads, rest 0  (32KB)
    __shared__ float xs[32 * 136];   // 32 j rows x 128 d cols, pad->136      (17KB)

    const int b  = blockIdx.x / 7;
    const int nc = blockIdx.x % 7;              // d-chunk == t slab
    const int tid = threadIdx.x;
    const int wv = tid >> 5, lane = tid & 31;
    const int half = lane >> 4, m = lane & 15;

    const float* wsm = W + (size_t)b * 4 * C_;
    #pragma unroll
    for (int i = 0; i < 32; ++i) {              // fill 8192 floats
        int idx = tid + i * 256;
        int j = idx >> 4, n = idx & 15;
        wt[idx] = (n < 4) ? wsm[n * C_ + j] : 0.f;
    }
    __syncthreads();

    v8f acc = {};
    const size_t xbase = (size_t)b * T_ * C_ * F_ + (size_t)nc * C_ * F_;

    for (int jc = 0; jc < 16; ++jc) {           // K chunks of 32 j
        const float4* src = (const float4*)(x + xbase + (size_t)(jc * 32) * F_);
        #pragma unroll
        for (int i = 0; i < 4; ++i) {           // 1024 float4
            int idx = tid + i * 256;
            int jj = idx >> 5, q = idx & 31;
            *(float4*)&xs[jj * 136 + q * 4] = src[(size_t)jj * 32 + q];
        }
        __syncthreads();

        #pragma unroll
        for (int k4 = 0; k4 < 8; ++k4) {
            int k0 = k4 * 4 + 2 * half;
            int jg = jc * 32 + k0;
            v2f a, bf;
            a.x  = wt[jg * 16 + m];             // A[m][k] = w16[m][j]
            a.y  = wt[(jg + 1) * 16 + m];
            bf.x = xs[k0 * 136 + wv * 16 + m];  // B[k][n] = xr[j][d]
            bf.y = xs[(k0 + 1) * 136 + wv * 16 + m];
            acc = __builtin_amdgcn_wmma_f32_16x16x4_f32(
                      false, a, false, bf, (short)0, acc, false, false);
        }
        __syncthreads();
    }

    if (half == 0) {                            // rows M=0..3 are the heads
        int d0 = nc * 128 + wv * 16;
        float* Yp = Y + (size_t)b * 4 * D_ + d0 + m;
        #pragma unroll
        for (int r = 0; r < 4; ++r) Yp[(size_t)r * D_] = acc[r];
    }
}

// ---------------- K4: vb[b][g] = y[b,h,:] @ W_heads[h][:,e] + Bg[g] ---------
__global__ void __launch_bounds__(256)
k4_v(const float* __restrict__ Wh, const float* __restrict__ Bg,
     const float* __restrict__ Y, float* __restrict__ VB)
{
    __shared__ float yl[4 * D_];                // 14KB
    const int b = blockIdx.x, tid = threadIdx.x;
    const float4* Ysrc = (const float4*)(Y + (size_t)b * 4 * D_);
    #pragma unroll
    for (int i = 0; i < 4; ++i) {
        int idx = tid + i * 256;
        if (idx < 896) *(float4*)&yl[idx * 4] = Ysrc[idx];
    }
    __syncthreads();

    for (int g = tid; g < G_; g += 256) {
        int h = g / HD_, e = g - h * HD_;
        const float* w = Wh + (size_t)h * D_ * HD_ + e;
        const float* y = yl + h * D_;
        float a0 = 0.f, a1 = 0.f, a2 = 0.f, a3 = 0.f;
        for (int d = 0; d < D_; d += 4) {
            a0 += y[d]     * w[(size_t)(d)     * HD_];
            a1 += y[d + 1] * w[(size_t)(d + 1) * HD_];
            a2 += y[d + 2] * w[(size_t)(d + 2) * HD_];
            a3 += y[d + 3] * w[(size_t)(d + 3) * HD_];
        }
        VB[(size_t)b * G_ + g] = Bg[g] + ((a0 + a1) + (a2 + a3));
    }
}

// ---------------- K5: out[b,t,c,f] = vb[b, t*128+f]  (broadcast store) ------
__global__ void __launch_bounds__(256)
k5_out(const float* __restrict__ VB, float* __restrict__ out)
{
    const int i  = blockIdx.x * 256 + threadIdx.x;  // float4 index
    const int fq = i & 31;
    const int tt = (i >> 14) % 7;
    const int b  = i / 114688;
    const float4* vb4 = (const float4*)VB;
    float4 v = vb4[(size_t)b * 224 + tt * 32 + fq];
    ((float4*)out)[i] = v;
}

// ---------------------------------------------------------------------------
extern "C" void kernel_launch(void* const* d_in, const int* in_sizes, int n_in,
                              void* d_out, int out_size, void* d_ws, size_t ws_size,
                              hipStream_t stream)
{
    const float* x  = (const float*)d_in[0];   // [64,7,512,128]
    const float* Wh = (const float*)d_in[1];   // [4,896,224]
    const float* Wa = (const float*)d_in[2];   // [4,448,1]
    const float* Bg = (const float*)d_in[3];   // [896]
    float* out = (float*)d_out;                // [64,7,512,128]
    float* ws  = (float*)d_ws;                 // needs >= 2.25 MB

    float* U  = ws + WS_U;
    float* S  = ws + WS_S;
    float* W  = ws + WS_W;
    float* Y  = ws + WS_Y;
    float* VB = ws + WS_VB;

    k0_u      <<<56,    256, 0, stream>>>(Wh, Wa, U);
    k1_scores <<<512,   128, 0, stream>>>(x, U, S);
    k2_softmax<<<32,    256, 0, stream>>>(S, W);
    k3_y      <<<448,   256, 0, stream>>>(x, W, Y);
    k4_v      <<<64,    256, 0, stream>>>(Wh, Bg, Y, VB);
    k5_out    <<<28672, 256, 0, stream>>>(VB, out);
}